// SS2D_17446157156534
// MI455X (gfx1250) — compile-verified
//
#include <hip/hip_runtime.h>
#include <hip/hip_bf16.h>

typedef float v2f __attribute__((ext_vector_type(2)));
typedef float v8f __attribute__((ext_vector_type(8)));

#define Bsz   2
#define Hdim  64
#define Wdim  64
#define Lsz   4096          // Hdim*Wdim
#define DM    96            // d_model
#define DD    192           // 2*d_model
#define NS    16            // d_state
#define RR    6             // dt rank
#define PP    38            // RR + 2*NS
#define PPAD  48            // padded proj stride
#define ROWS  (Bsz*Lsz)     // 8192

// ---------------- WMMA GEMM kernels (fp32, 16x16x4) ----------------
// A: row-major [ROWS x K], B: row-major [K x N], C: row-major [ROWS x ldc]
// ISA 7.12.2 layout: A 16x4: lanes 0-15 -> K={0,1}, lanes 16-31 -> K={2,3};
// B 4x16 mirrors (VGPR0: K=0/K=2 halves, VGPR1: K=1/K=3); C/D: VGPR v -> row v+8*hi.

__global__ void k_inproj(const float* __restrict__ x, const float* __restrict__ Win,
                         float* __restrict__ xz) {
    const int tilesN = 384 / 16;                 // 24
    int wave = blockIdx.x * (blockDim.x >> 5) + (threadIdx.x >> 5);
    int lane = threadIdx.x & 31;
    int tM = wave / tilesN, tN = wave % tilesN;
    int hi = lane >> 4, ln = lane & 15;
    int row = tM * 16 + ln;
    int col = tN * 16 + ln;
    v8f acc = {};
    for (int k0 = 0; k0 < DM; k0 += 4) {
        int ka = k0 + 2 * hi;
        v2f a;  a.x = x[row * DM + ka];        a.y = x[row * DM + ka + 1];
        v2f bf; bf.x = Win[ka * 384 + col];    bf.y = Win[(ka + 1) * 384 + col];
        acc = __builtin_amdgcn_wmma_f32_16x16x4_f32(false, a, false, bf,
                                                    (short)0, acc, false, false);
    }
#pragma unroll
    for (int v = 0; v < 8; ++v)
        xz[(size_t)(tM * 16 + v + 8 * hi) * 384 + col] = acc[v];
}

__global__ void k_xproj(const float* __restrict__ xc, const float* __restrict__ Wx,
                        float* __restrict__ proj) {
    const int tilesN = PPAD / 16;                // 3
    int wave = blockIdx.x * (blockDim.x >> 5) + (threadIdx.x >> 5);
    int lane = threadIdx.x & 31;
    int tM = wave / tilesN, tN = wave % tilesN;
    int hi = lane >> 4, ln = lane & 15;
    int row = tM * 16 + ln;
    int col = tN * 16 + ln;
    bool inb = (col < PP);
    v8f acc = {};
    for (int k0 = 0; k0 < DD; k0 += 4) {
        int ka = k0 + 2 * hi;
        v2f a;  a.x = xc[(size_t)row * DD + ka]; a.y = xc[(size_t)row * DD + ka + 1];
        v2f bf; bf.x = inb ? Wx[ka * PP + col] : 0.f;
                bf.y = inb ? Wx[(ka + 1) * PP + col] : 0.f;
        acc = __builtin_amdgcn_wmma_f32_16x16x4_f32(false, a, false, bf,
                                                    (short)0, acc, false, false);
    }
#pragma unroll
    for (int v = 0; v < 8; ++v)
        proj[(size_t)(tM * 16 + v + 8 * hi) * PPAD + col] = acc[v];
}

__global__ void k_outproj(const float* __restrict__ g, const float* __restrict__ Wout,
                          float* __restrict__ out) {
    const int tilesN = DM / 16;                  // 6
    int wave = blockIdx.x * (blockDim.x >> 5) + (threadIdx.x >> 5);
    int lane = threadIdx.x & 31;
    int tM = wave / tilesN, tN = wave % tilesN;
    int hi = lane >> 4, ln = lane & 15;
    int row = tM * 16 + ln;
    int col = tN * 16 + ln;
    v8f acc = {};
    for (int k0 = 0; k0 < DD; k0 += 4) {
        int ka = k0 + 2 * hi;
        v2f a;  a.x = g[(size_t)row * DD + ka];  a.y = g[(size_t)row * DD + ka + 1];
        v2f bf; bf.x = Wout[ka * DM + col];      bf.y = Wout[(ka + 1) * DM + col];
        acc = __builtin_amdgcn_wmma_f32_16x16x4_f32(false, a, false, bf,
                                                    (short)0, acc, false, false);
    }
#pragma unroll
    for (int v = 0; v < 8; ++v)
        out[(size_t)(tM * 16 + v + 8 * hi) * DM + col] = acc[v];
}

// ---------------- depthwise 3x3 conv + SiLU (channels-last) ----------------
__global__ void k_conv(const float* __restrict__ xz, const float* __restrict__ cw,
                       const float* __restrict__ cb, float* __restrict__ xc) {
    int i = blockIdx.x * blockDim.x + threadIdx.x;
    if (i >= ROWS * DD) return;
    int d = i % DD;
    int w = (i / DD) & 63;
    int h = (i / (DD * Wdim)) & 63;
    int b = i / (DD * Lsz);
    float acc = cb[d];
#pragma unroll
    for (int di = -1; di <= 1; ++di) {
        int hh = h + di; if (hh < 0 || hh > 63) continue;
#pragma unroll
        for (int dj = -1; dj <= 1; ++dj) {
            int ww = w + dj; if (ww < 0 || ww > 63) continue;
            float v = xz[((size_t)b * Lsz + hh * Wdim + ww) * 384 + d];  // xs = xz[...,0:192]
            acc = fmaf(v, cw[d * 9 + (di + 1) * 3 + (dj + 1)], acc);
        }
    }
    float sig = 1.f / (1.f + __expf(-acc));
    xc[i] = acc * sig;
}

// ---------------- selective scan: 16 lanes per (b,k,d) chain ----------------
__global__ void k_scan(const float* __restrict__ xc, const float* __restrict__ proj,
                       const float* __restrict__ W_dt, const float* __restrict__ b_dt,
                       const float* __restrict__ A_logs, const float* __restrict__ Ds,
                       float* __restrict__ ybuf) {
    int t = blockIdx.x * blockDim.x + threadIdx.x;   // 24576 threads
    int n = t & 15;                                  // state index (lane-parallel)
    int chain = t >> 4;                              // 0..1535
    int d = chain % DD;
    int k = (chain / DD) & 3;
    int b = chain / (DD * 4);

    float Wr[RR];
#pragma unroll
    for (int r = 0; r < RR; ++r) Wr[r] = W_dt[(k * RR + r) * DD + d];
    float bias = b_dt[k * DD + d];
    float Aval = -__expf(A_logs[((size_t)k * DD + d) * NS + n]);
    float Dval = Ds[k * DD + d];
    float s = 0.f;
    float* yk = ybuf + (size_t)k * (ROWS * DD);

    for (int l = 0; l < Lsz; ++l) {
        int lp  = (k & 1) ? (Lsz - 1 - l) : l;
        int pos = (k & 2) ? (((lp & 63) << 6) | (lp >> 6)) : lp;   // transpose map
        int base = b * Lsz + pos;
        const float* pr = proj + (size_t)base * PPAD;

        // prefetch ~24 steps ahead (transposed directions stride 64 rows/step)
        int lf = l + 24;
        if (lf < Lsz) {
            int lpf  = (k & 1) ? (Lsz - 1 - lf) : lf;
            int posf = (k & 2) ? (((lpf & 63) << 6) | (lpf >> 6)) : lpf;
            int basef = b * Lsz + posf;
            __builtin_prefetch(proj + (size_t)basef * PPAD, 0, 3);
            __builtin_prefetch(xc + (size_t)basef * DD + d, 0, 3);
        }

        float u = xc[(size_t)base * DD + d];
        float dtr = bias;
#pragma unroll
        for (int r = 0; r < RR; ++r) dtr = fmaf(pr[r], Wr[r], dtr);
        // softplus (stable): max(x,0) + log1p(exp(-|x|))
        float delta = fmaxf(dtr, 0.f) + log1pf(__expf(-fabsf(dtr)));

        float dA = __expf(delta * Aval);
        s = fmaf(dA, s, delta * pr[RR + n] * u);     // s = dA*s + delta*B_n*u
        float part = s * pr[RR + NS + n];            // s*C_n
#pragma unroll
        for (int off = 8; off > 0; off >>= 1)
            part += __shfl_xor(part, off, 16);
        if (n == 0)
            yk[(size_t)base * DD + d] = part + u * Dval;  // scatter at spatial pos
    }
}

// ---------------- combine 4 directions + SiLU(z) gate ----------------
__global__ void k_gate(const float* __restrict__ ybuf, const float* __restrict__ xz,
                       float* __restrict__ g) {
    int i = blockIdx.x * blockDim.x + threadIdx.x;
    if (i >= ROWS * DD) return;
    int d = i % DD;
    int base = i / DD;
    const size_t S = (size_t)ROWS * DD;
    float y = ybuf[i] + ybuf[S + i] + ybuf[2 * S + i] + ybuf[3 * S + i];
    float z = xz[(size_t)base * 384 + DD + d];
    float sig = 1.f / (1.f + __expf(-z));
    g[i] = y * (z * sig);
}

// ---------------- launcher ----------------
extern "C" void kernel_launch(void* const* d_in, const int* in_sizes, int n_in,
                              void* d_out, int out_size, void* d_ws, size_t ws_size,
                              hipStream_t stream) {
    const float* x      = (const float*)d_in[0];
    const float* W_in   = (const float*)d_in[1];
    const float* conv_w = (const float*)d_in[2];
    const float* conv_b = (const float*)d_in[3];
    const float* W_x    = (const float*)d_in[4];
    const float* W_dt   = (const float*)d_in[5];
    const float* b_dt   = (const float*)d_in[6];
    const float* A_logs = (const float*)d_in[7];
    const float* Ds     = (const float*)d_in[8];
    const float* W_out  = (const float*)d_in[9];
    float* out = (float*)d_out;

    float* ws = (float*)d_ws;
    const size_t OFF_XZ   = 0;                                    // 3,145,728
    const size_t OFF_XC   = OFF_XZ + (size_t)ROWS * 384;          // 1,572,864
    const size_t OFF_PROJ = OFF_XC + (size_t)ROWS * DD;           //   393,216
    const size_t OFF_Y    = OFF_PROJ + (size_t)ROWS * PPAD;       // 6,291,456
    const size_t OFF_G    = OFF_Y + 4 * (size_t)ROWS * DD;        // 1,572,864
    float* xzb  = ws + OFF_XZ;
    float* xcb  = ws + OFF_XC;
    float* prj  = ws + OFF_PROJ;
    float* ybuf = ws + OFF_Y;
    float* gbuf = ws + OFF_G;

    // 1) xz = x @ W_in : 512x24 tiles, 4 waves per block
    k_inproj<<<dim3((512 * 24) / 4), dim3(128), 0, stream>>>(x, W_in, xzb);
    // 2) depthwise conv + SiLU
    k_conv<<<dim3((ROWS * DD) / 256), dim3(256), 0, stream>>>(xzb, conv_w, conv_b, xcb);
    // 3) proj = xc @ W_x : 512x3 tiles
    k_xproj<<<dim3((512 * 3) / 4), dim3(128), 0, stream>>>(xcb, W_x, prj);
    // 4) 4-direction selective scan (1536 chains x 16 lanes)
    k_scan<<<dim3((1536 * 16) / 256), dim3(256), 0, stream>>>(xcb, prj, W_dt, b_dt,
                                                              A_logs, Ds, ybuf);
    // 5) combine + gate
    k_gate<<<dim3((ROWS * DD) / 256), dim3(256), 0, stream>>>(ybuf, xzb, gbuf);
    // 6) out = g @ W_out : 512x6 tiles
    k_outproj<<<dim3((512 * 6) / 4), dim3(128), 0, stream>>>(gbuf, W_out, out);
}